// ToposYonedaModel_9921374454412
// MI455X (gfx1250) — compile-verified
//
#include <hip/hip_runtime.h>
#include <stdint.h>

// sigmoid(M[u,v]) gather kernel for MI455X (gfx1250, wave32).
//
// Memory-bound random gather: ~120 MB of unique cachelines per pass, which
// fits the 192 MB L2 -> keep matrix loads TEMPORAL (default RT) so graph
// replays hit L2. Output is write-once -> non-temporal store. Index streams
// are contiguous -> b128 loads + global_prefetch_b8 run-ahead.

typedef float  f32x4 __attribute__((ext_vector_type(4)));
typedef int    i32x4 __attribute__((ext_vector_type(4)));

#define LOG2E 1.4426950408889634074f

__device__ __forceinline__ float sigmoid_f32(float x) {
  // sigmoid(x) = 1 / (1 + exp(-x)), exp via hardware v_exp_f32 (base-2).
  float e = __builtin_amdgcn_exp2f(-x * LOG2E);
  return 1.0f / (1.0f + e);
}

__global__ void __launch_bounds__(256)
topos_yoneda_gather_sigmoid(const float* __restrict__ M,   // [16384, 16384]
                            const int*   __restrict__ u,   // [n]
                            const int*   __restrict__ v,   // [n]
                            float*       __restrict__ out, // [n]
                            int n) {
  const int quad = blockIdx.x * blockDim.x + threadIdx.x;
  const int base = quad * 4;

  if (base + 4 <= n) {
    // Run the contiguous index streams ahead of the wavefront.
    // Speculative prefetch: OOB translations are silently dropped (ISA 7.3).
    __builtin_prefetch(u + base + 4096, 0, 1);   // global_prefetch_b8
    __builtin_prefetch(v + base + 4096, 0, 1);

    // Coalesced 128-bit index loads (global_load_b128).
    i32x4 u4 = *reinterpret_cast<const i32x4*>(u + base);
    i32x4 v4 = *reinterpret_cast<const i32x4*>(v + base);

    // Element offsets: u*16384 + v < 2^28; byte offset < 2^30 -> fits the
    // 32-bit VGPR offset of the GVS global addressing mode.
    uint32_t o0 = ((uint32_t)u4.x << 14) + (uint32_t)v4.x;
    uint32_t o1 = ((uint32_t)u4.y << 14) + (uint32_t)v4.y;
    uint32_t o2 = ((uint32_t)u4.z << 14) + (uint32_t)v4.z;
    uint32_t o3 = ((uint32_t)u4.w << 14) + (uint32_t)v4.w;

    // Four independent gathers in flight per lane before first use (MLP).
    // Default temporal policy (RT): gathered lines stay in the 192 MB L2,
    // so replays of the identical gather become largely L2-resident.
    float x0 = M[o0];
    float x1 = M[o1];
    float x2 = M[o2];
    float x3 = M[o3];

    f32x4 r;
    r.x = sigmoid_f32(x0);
    r.y = sigmoid_f32(x1);
    r.z = sigmoid_f32(x2);
    r.w = sigmoid_f32(x3);

    // Write-once output: non-temporal store keeps 4 MB out of L2, leaving
    // more room for the gather working set.
    __builtin_nontemporal_store(r, reinterpret_cast<f32x4*>(out + base));
  } else if (base < n) {
    // Tail (unused for n = 1048576, kept for generality).
    for (int i = base; i < n; ++i) {
      uint32_t o = ((uint32_t)u[i] << 14) + (uint32_t)v[i];
      out[i] = sigmoid_f32(M[o]);
    }
  }
}

extern "C" void kernel_launch(void* const* d_in, const int* in_sizes, int n_in,
                              void* d_out, int out_size, void* d_ws, size_t ws_size,
                              hipStream_t stream) {
  const float* M   = (const float*)d_in[0];  // morphisms_logits, f32 [16384*16384]
  const int*   u   = (const int*)d_in[1];    // i32 [N_PAIRS]
  const int*   v   = (const int*)d_in[2];    // i32 [N_PAIRS]
  float*       out = (float*)d_out;          // f32 [N_PAIRS]

  const int n       = in_sizes[1];           // 1048576
  const int quads   = (n + 3) / 4;
  const int threads = 256;                   // 8 wave32 waves per block
  const int blocks  = (quads + threads - 1) / threads;

  topos_yoneda_gather_sigmoid<<<blocks, threads, 0, stream>>>(M, u, v, out, n);
}